// LowRankRayTracer_89824946028957
// MI455X (gfx1250) — compile-verified
//
#include <hip/hip_runtime.h>

typedef __attribute__((ext_vector_type(2))) float v2f;
typedef __attribute__((ext_vector_type(4))) float v4f;
typedef __attribute__((ext_vector_type(8))) float v8f;

#define NUM_D 1024  // directions
#define NUM_P 256   // sampling points
#define NUM_R 64    // rank
#define NUM_K 1024  // subcarriers

// ---------------------------------------------------------------------------
// Kernel 1: stream the two (D,P,R) tensors once and reduce over P.
//   U[d,r] = sum_p atten[d,p,r],  V[d,r] = sum_p rad[d,p,r]
// One block per direction d. 256 threads: thread t owns r-group rg = t%16
// (4 consecutive r via v4f / b128 loads) and p-slot pg = t/16 (16-way
// partial over p). Non-temporal loads: data is touched exactly once.
// ---------------------------------------------------------------------------
__global__ __launch_bounds__(256) void lrrt_reduce_p(
    const float* __restrict__ att, const float* __restrict__ rad,
    float* __restrict__ U, float* __restrict__ V) {
  __shared__ v4f s[256];

  const int d  = blockIdx.x;
  const int t  = threadIdx.x;
  const int rg = t & 15;   // which group of 4 r-values
  const int pg = t >> 4;   // which p partial slot (0..15)

  const size_t base = (size_t)d * NUM_P * NUM_R + (size_t)rg * 4;

  v4f accA = {};
  v4f accB = {};

#pragma unroll
  for (int p = 0; p < NUM_P; p += 16) {
    const size_t idx = base + (size_t)(p + pg) * NUM_R;
    const v4f a = __builtin_nontemporal_load(
        reinterpret_cast<const v4f*>(att + idx));
    const v4f b = __builtin_nontemporal_load(
        reinterpret_cast<const v4f*>(rad + idx));
    accA += a;
    accB += b;
  }

  // ---- reduce attenuation partials over pg, write U[d, rg*4 .. rg*4+3] ----
  s[t] = accA;
  __syncthreads();
#pragma unroll
  for (int off = 8; off > 0; off >>= 1) {
    if (pg < off) {
      s[pg * 16 + rg] += s[(pg + off) * 16 + rg];
    }
    __syncthreads();
  }
  if (pg == 0) {
    *reinterpret_cast<v4f*>(U + (size_t)d * NUM_R + rg * 4) = s[rg];
  }
  __syncthreads();

  // ---- same for radiation partials -> V ----
  s[t] = accB;
  __syncthreads();
#pragma unroll
  for (int off = 8; off > 0; off >>= 1) {
    if (pg < off) {
      s[pg * 16 + rg] += s[(pg + off) * 16 + rg];
    }
    __syncthreads();
  }
  if (pg == 0) {
    *reinterpret_cast<v4f*>(V + (size_t)d * NUM_R + rg * 4) = s[rg];
  }
}

// ---------------------------------------------------------------------------
// Kernel 2: fused WMMA GEMMs + d-reduction.
//   csi[k] = (1/D) * sum_d (U[d,:]·F[k,:]) * (V[d,:]·F[k,:])
// Grid: 64 blocks, one per 16-wide k-tile. 256 threads = 8 wave32 waves.
// Per wave: keep the F-tile as 16 B-fragments (K=4 steps over R=64) in
// registers; loop over 8 of the 64 d-tiles. For each d-tile, preload ALL 32
// A-fragments (atten + rad); sched_barrier(0) pins the schedule so the
// backend issues the whole load batch (clauses, all outstanding) before a
// single wait, then runs the two V_WMMA_F32_16X16X4_F32 accumulation chains
// uninterrupted. Elementwise product of the two 16x16 accumulator tiles is
// reduced over d (registers), lanes (shfl), and waves (LDS) — no atomics,
// fully deterministic.
//
// Fragment indexing (wave32):
//   A 16x4 f32: lane l -> row m = l%16, VGPR pair holds K = 2*(l/16)+{0,1}
//   B 4x16 f32: lane l -> col n = l%16, VGPR pair holds K = 2*(l/16)+{0,1}
//   C/D 16x16:  VGPR j, lanes 0-15 -> (M=j, N=lane); lanes 16-31 -> (M=j+8)
// ---------------------------------------------------------------------------
__global__ __launch_bounds__(256) void lrrt_wmma_csi(
    const float* __restrict__ U, const float* __restrict__ V,
    const float* __restrict__ F, float* __restrict__ out) {
  __shared__ float sred[8][16];

  const int ktile = blockIdx.x;          // 0..63
  const int wave  = threadIdx.x >> 5;    // 0..7
  const int lane  = threadIdx.x & 31;
  const int n     = lane & 15;           // column (k) / row (d) within tile
  const int half  = lane >> 4;           // K-pair select within a 4-step

  // Load the B-matrix (frequency basis) fragments for this k-tile once.
  v2f bfrag[16];
  const float* frow = F + (size_t)(ktile * 16 + n) * NUM_R + half * 2;
#pragma unroll
  for (int s = 0; s < 16; ++s) {
    bfrag[s] = *reinterpret_cast<const v2f*>(frow + s * 4);
  }

  float partial = 0.f;

  // Each wave handles d-tiles wave, wave+8, ..., wave+56 (8 tiles).
  for (int dt = wave; dt < NUM_D / 16; dt += 8) {
    const float* urow = U + (size_t)(dt * 16 + n) * NUM_R + half * 2;
    const float* vrow = V + (size_t)(dt * 16 + n) * NUM_R + half * 2;

    // Preload all A-fragments for this d-tile: 32 outstanding b64 loads.
    v2f aufrag[16];
    v2f avfrag[16];
#pragma unroll
    for (int s = 0; s < 16; ++s) {
      aufrag[s] = *reinterpret_cast<const v2f*>(urow + s * 4);
      avfrag[s] = *reinterpret_cast<const v2f*>(vrow + s * 4);
    }

    // Hard scheduling fence: nothing crosses. Forces the load batch above
    // to be issued (and waited once) before the WMMA chain below — prevents
    // the scheduler from re-serializing into load->wait->wmma per step.
    __builtin_amdgcn_sched_barrier(0);

    v8f accA = {};
    v8f accB = {};
#pragma unroll
    for (int s = 0; s < 16; ++s) {
      accA = __builtin_amdgcn_wmma_f32_16x16x4_f32(
          false, aufrag[s], false, bfrag[s], (short)0, accA, false, false);
      accB = __builtin_amdgcn_wmma_f32_16x16x4_f32(
          false, avfrag[s], false, bfrag[s], (short)0, accB, false, false);
    }

    __builtin_amdgcn_sched_barrier(0);

    // csi contribution: elementwise product of the two factor tiles,
    // summed over the 8 rows this lane holds.
#pragma unroll
    for (int j = 0; j < 8; ++j) partial += accA[j] * accB[j];
  }

  // Fold lanes 16..31 (rows M=8..15) onto lanes 0..15 (wave32 shuffle).
  partial += __shfl_down(partial, 16);
  if (lane < 16) sred[wave][n] = partial;
  __syncthreads();

  if (threadIdx.x < 16) {
    float sum = 0.f;
#pragma unroll
    for (int w = 0; w < 8; ++w) sum += sred[w][threadIdx.x];
    out[ktile * 16 + threadIdx.x] = sum * (1.0f / (float)NUM_D);
  }
}

// ---------------------------------------------------------------------------
extern "C" void kernel_launch(void* const* d_in, const int* in_sizes, int n_in,
                              void* d_out, int out_size, void* d_ws,
                              size_t ws_size, hipStream_t stream) {
  const float* att = (const float*)d_in[0];  // (D,P,R) f32
  const float* rad = (const float*)d_in[1];  // (D,P,R) f32
  const float* F   = (const float*)d_in[2];  // (K,R)  f32
  float* out = (float*)d_out;                // (K,)   f32

  float* U = (float*)d_ws;                   // (D,R) f32 = 256 KB
  float* V = U + (size_t)NUM_D * NUM_R;      // (D,R) f32 = 256 KB

  lrrt_reduce_p<<<NUM_D, 256, 0, stream>>>(att, rad, U, V);
  lrrt_wmma_csi<<<NUM_K / 16, 256, 0, stream>>>(U, V, F, out);
}